// MyRGCNConvNaive_35536559407732
// MI455X (gfx1250) — compile-verified
//
#include <hip/hip_runtime.h>
#include <hip/hip_bf16.h>

#define N_NODE 50000
#define N_EDGE 1000000
#define IN_CH  64
#define HID_CH 64
#define NUM_REL 8

typedef __attribute__((ext_vector_type(2))) float v2f;
typedef __attribute__((ext_vector_type(8))) float v8f;

// ---------------------------------------------------------------------------
// Kernel 0: zero the output accumulator (harness poisons d_out with 0xAA).
// 50000*64 = 3,200,000 floats = 800,000 float4. 3125 blocks * 256 threads.
// ---------------------------------------------------------------------------
__global__ void rgcn_zero_out(float4* __restrict__ out4) {
    unsigned i = blockIdx.x * blockDim.x + threadIdx.x;
    if (i < (N_NODE * HID_CH) / 4) {
        out4[i] = make_float4(0.f, 0.f, 0.f, 0.f);
    }
}

// ---------------------------------------------------------------------------
// Kernel 1: xw[r] = x @ W[r] for all 8 relations, via V_WMMA_F32_16X16X4_F32.
// Grid: (3125 row-tiles, 8 relations). Block: 128 threads = 4 waves.
// Wave w computes the 16x16 tile at rows [16*bx,16*bx+16), cols [16*w,16*w+16).
// K = 64, stepped by 4 -> 16 WMMA ops per wave.
//
// A-fragment (16x4 f32, 2 VGPRs): lanes 0-15 hold row M=lane, K = k+0,k+1;
// lanes 16-31 hold row M=lane-16, K = k+2,k+3  => contiguous float2 per lane.
// B-fragment (4x16 f32, 2 VGPRs): VGPR0 = rows k+0 (lanes 0-15) / k+2
// (lanes 16-31); VGPR1 = rows k+1 / k+3, column N = n0 + (lane&15).
// C/D (16x16 f32, 8 VGPRs): VGPR i -> row M=i (lanes 0-15), M=i+8 (lanes 16-31).
// EXEC is all-1s: no divergence anywhere in this kernel.
// ---------------------------------------------------------------------------
__global__ void __launch_bounds__(128)
rgcn_gemm_wmma(const float* __restrict__ x,       // [N_NODE, IN_CH]
               const float* __restrict__ Wmat,    // [NUM_REL, IN_CH, HID_CH]
               float* __restrict__ xw)            // [NUM_REL, N_NODE, HID_CH]
{
    const int lane   = threadIdx.x & 31;
    const int wave   = threadIdx.x >> 5;       // 0..3 -> N-tile
    const int row0   = blockIdx.x << 4;        // 16-row tile (3125*16 == 50000)
    const int n0     = wave << 4;              // 16-col tile
    const int r      = blockIdx.y;             // relation

    const float* __restrict__ Wr = Wmat + (size_t)r * IN_CH * HID_CH;

    const int laneLo = lane & 15;              // row within half-wave
    const int laneHi = lane >> 4;              // 0 or 1 (half-wave select)

    // A addressing: row = row0 + laneLo, col = k + 2*laneHi (+0,+1)
    const float* __restrict__ aPtr = x + (size_t)(row0 + laneLo) * IN_CH + (laneHi << 1);
    // B addressing: row = k + 2*laneHi (+0,+1), col = n0 + laneLo
    const float* __restrict__ bPtr = Wr + (size_t)(laneHi << 1) * HID_CH + n0 + laneLo;

    v8f acc = {};
    #pragma unroll
    for (int k = 0; k < IN_CH; k += 4) {
        const float2 a2 = *(const float2*)(aPtr + k);
        v2f A; A.x = a2.x; A.y = a2.y;
        v2f B;
        B.x = bPtr[(size_t)k * HID_CH];
        B.y = bPtr[(size_t)(k + 1) * HID_CH];
        // 8 args: (neg_a, A, neg_b, B, c_mod, C, reuse_a, reuse_b)
        acc = __builtin_amdgcn_wmma_f32_16x16x4_f32(
            /*neg_a=*/false, A, /*neg_b=*/false, B,
            /*c_mod=*/(short)0, acc, /*reuse_a=*/false, /*reuse_b=*/false);
    }

    // Store D: VGPR i holds (row0 + i + 8*laneHi, n0 + laneLo)
    float* __restrict__ outBase =
        xw + ((size_t)r * N_NODE + (size_t)(row0 + (laneHi << 3))) * HID_CH + n0 + laneLo;
    #pragma unroll
    for (int i = 0; i < 8; ++i) {
        outBase[(size_t)i * HID_CH] = acc[i];
    }
}

// ---------------------------------------------------------------------------
// Kernel 2: per edge e: out[dst[e]] += xw[rel[e]][src[e]].
// One wave per edge: 32 lanes x float2 = 64 channels. xw (102.4 MB) is
// L2-resident on MI455X (192 MB L2), so the random gathers mostly hit L2.
// unsafeAtomicAdd -> GLOBAL_ATOMIC_ADD_F32 (no return, STOREcnt tracked).
// ---------------------------------------------------------------------------
__global__ void __launch_bounds__(256)
rgcn_scatter(const float* __restrict__ xw,       // [NUM_REL, N_NODE, HID_CH]
             const int* __restrict__ edge_index, // [2, N_EDGE] int32
             const int* __restrict__ rel,        // [N_EDGE] int32
             float* __restrict__ out)            // [N_NODE, HID_CH]
{
    const int lane = threadIdx.x & 31;
    const int wave = threadIdx.x >> 5;                 // 0..7
    const int e    = (blockIdx.x << 3) + wave;
    if (e >= N_EDGE) return;

    const int s = edge_index[e];                       // src row
    const int d = edge_index[N_EDGE + e];              // dst row
    const int r = rel[e];

    const float2 v = *(const float2*)(
        xw + (((size_t)r * N_NODE + (size_t)s) * HID_CH) + (lane << 1));

    float* __restrict__ o = out + (size_t)d * HID_CH + (lane << 1);
#if defined(__HIP_DEVICE_COMPILE__)
    unsafeAtomicAdd(o,     v.x);
    unsafeAtomicAdd(o + 1, v.y);
#else
    atomicAdd(o,     v.x);
    atomicAdd(o + 1, v.y);
#endif
}

// ---------------------------------------------------------------------------
// Host launcher. Inputs (setup_inputs order):
//   d_in[0] x          : float32 [50000,64]
//   d_in[1] W          : float32 [8,64,64]
//   d_in[2] edge_index : int32   [2,1000000]
//   d_in[3] rel        : int32   [1000000]
//   d_in[4] num_node (1), d_in[5] num_edge (1) -- compile-time constants here.
// d_ws holds xw: 8*50000*64 floats = 102.4 MB.
// ---------------------------------------------------------------------------
extern "C" void kernel_launch(void* const* d_in, const int* in_sizes, int n_in,
                              void* d_out, int out_size, void* d_ws, size_t ws_size,
                              hipStream_t stream) {
    const float* x          = (const float*)d_in[0];
    const float* Wmat       = (const float*)d_in[1];
    const int*   edge_index = (const int*)d_in[2];
    const int*   rel        = (const int*)d_in[3];
    float*       out        = (float*)d_out;
    float*       xw         = (float*)d_ws;

    // 1) zero the output accumulator
    rgcn_zero_out<<<(N_NODE * HID_CH / 4 + 255) / 256, 256, 0, stream>>>(
        (float4*)out);

    // 2) xw[r] = x @ W[r]  (WMMA f32 16x16x4)
    rgcn_gemm_wmma<<<dim3(N_NODE / 16, NUM_REL, 1), 128, 0, stream>>>(
        x, Wmat, xw);

    // 3) gather + scatter-add over edges
    rgcn_scatter<<<(N_EDGE + 7) / 8, 256, 0, stream>>>(
        xw, edge_index, rel, out);
}